// MagDetNet_42786464202768
// MI455X (gfx1250) — compile-verified
//
#include <hip/hip_runtime.h>
#include <hip/hip_bf16.h>
#include <math.h>

typedef __attribute__((ext_vector_type(2))) float v2f;
typedef __attribute__((ext_vector_type(8))) float v8f;

#define B_     512
#define S_     264
#define BL_    256
#define V_     128
#define Z_     1024
#define NOH_   1056   // S*OH
#define KIN_   392    // S+V
#define TWO_S  528
#define TWO_BL 512

// ---------------------------------------------------------------------------
// Kernel 1: stream Psi once per (batch, e/o); fused Px, u, and Psi^T*w pass.
// contrib[psi][b][s] = cos(ph)*t[s] + sin(ph)*t[S+s]  (== (A@u)[b,s])
// ---------------------------------------------------------------------------
__global__ __launch_bounds__(256) void psi_pass_kernel(
    const float* __restrict__ x_mag, const float* __restrict__ x_phase,
    const float* __restrict__ y_e,   const float* __restrict__ y_o,
    const float* __restrict__ Psi_e, const float* __restrict__ Psi_o,
    const float* __restrict__ d1p, const float* __restrict__ d2p,
    const float* __restrict__ d4p,
    float* __restrict__ contrib)
{
    const int b    = blockIdx.x;
    const int psi  = blockIdx.y;
    const int tid  = threadIdx.x;
    const int lane = tid & 31;
    const int wave = tid >> 5;

    const float* __restrict__ Psi = psi ? Psi_o : Psi_e;
    const float* __restrict__ y   = psi ? y_o   : y_e;
    const float alpha = psi ? -d2p[0] : -d1p[0];
    const float beta  = psi ?  d4p[0] :  d2p[0];

    // per-lane x columns: col = j*128 + lane*4 + e ; x = [mag*cos, mag*sin]
    float xr[5][4];
#pragma unroll
    for (int j = 0; j < 5; ++j) {
#pragma unroll
        for (int e = 0; e < 4; ++e) {
            const int col = j * 128 + lane * 4 + e;
            float xv = 0.f;
            if (col < TWO_S) {
                const int idx = (col < S_) ? col : (col - S_);
                const float m = x_mag[b * S_ + idx];
                const float p = x_phase[b * S_ + idx];
                xv = m * ((col < S_) ? __cosf(p) : __sinf(p));
            }
            xr[j][e] = xv;
        }
    }

    float tacc[5][4];
#pragma unroll
    for (int j = 0; j < 5; ++j)
#pragma unroll
        for (int e = 0; e < 4; ++e) tacc[j][e] = 0.f;

    const float* __restrict__ Pb = Psi + (size_t)b * TWO_BL * TWO_S;
    const float* __restrict__ yb = y + b * BL_;

    for (int i = 0; i < 32; ++i) {
        const int m = wave * 32 + i;  // row-pair index, 8 waves cover BL=256
        const float* __restrict__ rt = Pb + (size_t)m * TWO_S;
        const float* __restrict__ rb = Pb + (size_t)(m + BL_) * TWO_S;
        if (i + 1 < 32) {  // pull next row pair toward the caches
            __builtin_prefetch(rt + TWO_S + lane * 16, 0, 0);
            __builtin_prefetch(rb + TWO_S + lane * 16, 0, 0);
        }

        float top[5][4], bot[5][4];
#pragma unroll
        for (int j = 0; j < 5; ++j) {
            const int c0 = j * 128 + lane * 4;
            if (c0 + 3 < TWO_S) {
                const float4 t4 = *(const float4*)(rt + c0);
                const float4 b4 = *(const float4*)(rb + c0);
                top[j][0] = t4.x; top[j][1] = t4.y; top[j][2] = t4.z; top[j][3] = t4.w;
                bot[j][0] = b4.x; bot[j][1] = b4.y; bot[j][2] = b4.z; bot[j][3] = b4.w;
            } else {
#pragma unroll
                for (int e = 0; e < 4; ++e) { top[j][e] = 0.f; bot[j][e] = 0.f; }
            }
        }

        float da = 0.f, db = 0.f;
#pragma unroll
        for (int j = 0; j < 5; ++j)
#pragma unroll
            for (int e = 0; e < 4; ++e) {
                da = fmaf(top[j][e], xr[j][e], da);
                db = fmaf(bot[j][e], xr[j][e], db);
            }
#pragma unroll
        for (int off = 16; off > 0; off >>= 1) {
            da += __shfl_xor(da, off, 32);
            db += __shfl_xor(db, off, 32);
        }

        const float u  = fmaf(beta, fmaf(da, da, db * db), alpha * yb[m]);
        const float wt = da * u;
        const float wb = db * u;
#pragma unroll
        for (int j = 0; j < 5; ++j)
#pragma unroll
            for (int e = 0; e < 4; ++e)
                tacc[j][e] = fmaf(wt, top[j][e], fmaf(wb, bot[j][e], tacc[j][e]));
    }

    // merge per-wave t partials (lane ownership is disjoint within a wave)
    __shared__ float t_sh[TWO_S];
    for (int idx = tid; idx < TWO_S; idx += 256) t_sh[idx] = 0.f;
    __syncthreads();
#pragma unroll
    for (int j = 0; j < 5; ++j)
#pragma unroll
        for (int e = 0; e < 4; ++e) {
            const int col = j * 128 + lane * 4 + e;
            if (col < TWO_S) atomicAdd(&t_sh[col], tacc[j][e]);
        }
    __syncthreads();

    if (tid < S_) {
        const float p = x_phase[b * S_ + tid];
        contrib[((size_t)psi * B_ + b) * S_ + tid] =
            __cosf(p) * t_sh[tid] + __sinf(p) * t_sh[S_ + tid];
    }
}

// ---------------------------------------------------------------------------
// Kernel 2: q = mag + contrib_e + contrib_o ; zin = [q, v]; copy old frame
// ---------------------------------------------------------------------------
__global__ __launch_bounds__(512) void assemble_kernel(
    const float* __restrict__ x_mag, const float* __restrict__ v,
    const float* __restrict__ contrib,
    float* __restrict__ zin, float* __restrict__ out_frame0)
{
    const int b = blockIdx.x, tid = threadIdx.x;
    if (tid < S_) {
        const float xm = x_mag[b * S_ + tid];
        out_frame0[b * S_ + tid] = xm;
        zin[b * KIN_ + tid] = xm + contrib[(size_t)b * S_ + tid]
                                 + contrib[((size_t)B_ + b) * S_ + tid];
    } else if (tid < KIN_) {
        zin[b * KIN_ + tid] = v[b * V_ + (tid - S_)];
    }
}

// ---------------------------------------------------------------------------
// fp32 WMMA GEMMs: each wave owns one 16x16 tile, K in steps of 4.
// A frag: lanes0-15 row m0+l, K=kk+{0,1}; lanes16-31 same rows, K=kk+{2,3}.
// B frag mirrors with N across lanes.  D VGPR r: row m0+r(+8 for hi lanes).
// ---------------------------------------------------------------------------
__global__ __launch_bounds__(256) void gemm1_kernel(
    const float* __restrict__ zin, const float* __restrict__ W1,
    const float* __restrict__ b1, float* __restrict__ z)
{
    const int tid = threadIdx.x, lane = tid & 31, wave = tid >> 5;
    const int m0 = blockIdx.x * 32 + (wave >> 2) * 16;
    const int n0 = blockIdx.y * 64 + (wave & 3) * 16;
    const int l15 = lane & 15, hi = lane >> 4;
    const int arow = m0 + l15, bcol = n0 + l15;

    v8f acc = {};
    for (int kk = 0; kk < KIN_; kk += 4) {
        const int kb = kk + 2 * hi;
        v2f a, bb;
        a.x  = zin[arow * KIN_ + kb];  a.y  = zin[arow * KIN_ + kb + 1];
        bb.x = W1[bcol * KIN_ + kb];   bb.y = W1[bcol * KIN_ + kb + 1];
        acc = __builtin_amdgcn_wmma_f32_16x16x4_f32(
            false, a, false, bb, (short)0, acc, false, false);
    }
#pragma unroll
    for (int r = 0; r < 8; ++r) {
        const int row = m0 + r + 8 * hi;
        const int col = n0 + l15;
        const float val = acc[r] + b1[col];
        z[row * Z_ + col] = val > 0.f ? val : 0.f;
    }
}

__global__ __launch_bounds__(256) void gemm2_kernel(
    const float* __restrict__ z, const float* __restrict__ W2,
    const float* __restrict__ b2, const float* __restrict__ x_mag_oh,
    float* __restrict__ out_oh, float* __restrict__ out_sym)
{
    const int tid = threadIdx.x, lane = tid & 31, wave = tid >> 5;
    const int m0 = blockIdx.x * 32 + (wave >> 2) * 16;
    const int n0 = blockIdx.y * 64 + (wave & 3) * 16;
    if (n0 >= NOH_) return;  // wave-uniform
    const int l15 = lane & 15, hi = lane >> 4;
    const int arow = m0 + l15, bcol = n0 + l15;

    v8f acc = {};
    for (int kk = 0; kk < Z_; kk += 4) {
        const int kb = kk + 2 * hi;
        v2f a, bb;
        a.x  = z[arow * Z_ + kb];   a.y  = z[arow * Z_ + kb + 1];
        bb.x = W2[bcol * Z_ + kb];  bb.y = W2[bcol * Z_ + kb + 1];
        acc = __builtin_amdgcn_wmma_f32_16x16x4_f32(
            false, a, false, bb, (short)0, acc, false, false);
    }
#pragma unroll
    for (int r = 0; r < 8; ++r) {
        const int row = m0 + r + 8 * hi;
        const int col = n0 + l15;
        const float val = acc[r] + b2[col] + x_mag_oh[row * NOH_ + col];
        out_oh[row * NOH_ + col] = val;
        // softmax over OH=4 group (lanes col&3 share a group), dot with MAPP
        float mx = val;
        mx = fmaxf(mx, __shfl_xor(mx, 1, 32));
        mx = fmaxf(mx, __shfl_xor(mx, 2, 32));
        float p = __expf(val - mx);
        float sum = p;
        sum += __shfl_xor(sum, 1, 32);
        sum += __shfl_xor(sum, 2, 32);
        float wv = (p / sum) * (0.5f * (float)((col & 3) + 1));  // MAPP
        wv += __shfl_xor(wv, 1, 32);
        wv += __shfl_xor(wv, 2, 32);
        if ((lane & 3) == 0) out_sym[row * S_ + (col >> 2)] = wv;
    }
}

__global__ __launch_bounds__(256) void gemm3_kernel(
    const float* __restrict__ z, const float* __restrict__ W3,
    const float* __restrict__ b3, const float* __restrict__ v,
    float* __restrict__ out_v)
{
    const int tid = threadIdx.x, lane = tid & 31, wave = tid >> 5;
    const int m0 = blockIdx.x * 32 + (wave >> 2) * 16;
    const int n0 = blockIdx.y * 64 + (wave & 3) * 16;
    const int l15 = lane & 15, hi = lane >> 4;
    const int arow = m0 + l15, bcol = n0 + l15;

    v8f acc = {};
    for (int kk = 0; kk < Z_; kk += 4) {
        const int kb = kk + 2 * hi;
        v2f a, bb;
        a.x  = z[arow * Z_ + kb];   a.y  = z[arow * Z_ + kb + 1];
        bb.x = W3[bcol * Z_ + kb];  bb.y = W3[bcol * Z_ + kb + 1];
        acc = __builtin_amdgcn_wmma_f32_16x16x4_f32(
            false, a, false, bb, (short)0, acc, false, false);
    }
#pragma unroll
    for (int r = 0; r < 8; ++r) {
        const int row = m0 + r + 8 * hi;
        const int col = n0 + l15;
        out_v[row * V_ + col] = acc[r] + b3[col] + v[row * V_ + col];
    }
}

// ---------------------------------------------------------------------------
extern "C" void kernel_launch(void* const* d_in, const int* in_sizes, int n_in,
                              void* d_out, int out_size, void* d_ws, size_t ws_size,
                              hipStream_t stream) {
    const float* x_mag    = (const float*)d_in[1];
    const float* x_mag_oh = (const float*)d_in[2];
    const float* v        = (const float*)d_in[3];
    const float* x_phase  = (const float*)d_in[4];
    const float* y_e      = (const float*)d_in[5];
    const float* y_o      = (const float*)d_in[6];
    const float* Psi_e    = (const float*)d_in[7];
    const float* Psi_o    = (const float*)d_in[8];
    const float* W1       = (const float*)d_in[9];
    const float* b1       = (const float*)d_in[10];
    const float* W2       = (const float*)d_in[11];
    const float* b2       = (const float*)d_in[12];
    const float* W3       = (const float*)d_in[13];
    const float* b3       = (const float*)d_in[14];
    const float* d1       = (const float*)d_in[15];
    const float* d2       = (const float*)d_in[16];
    const float* d4       = (const float*)d_in[18];  // d3 unused by reference

    float* ws      = (float*)d_ws;
    float* contrib = ws;                         // 2*B*S     = 270336
    float* zin     = contrib + 2 * B_ * S_;      // B*392     = 200704
    float* zact    = zin + B_ * KIN_;            // B*1024    = 524288

    float* out      = (float*)d_out;
    float* out_f0   = out;                               // old frame, B*S
    float* out_sym  = out + B_ * S_;                     // new frame, B*S
    float* out_oh   = out + 2 * B_ * S_;                 // B*1056
    float* out_v    = out + 2 * B_ * S_ + B_ * NOH_;     // B*128

    psi_pass_kernel<<<dim3(B_, 2), 256, 0, stream>>>(
        x_mag, x_phase, y_e, y_o, Psi_e, Psi_o, d1, d2, d4, contrib);
    assemble_kernel<<<dim3(B_), 512, 0, stream>>>(x_mag, v, contrib, zin, out_f0);
    gemm1_kernel<<<dim3(16, 16), 256, 0, stream>>>(zin, W1, b1, zact);
    gemm2_kernel<<<dim3(16, 17), 256, 0, stream>>>(zact, W2, b2, x_mag_oh, out_oh, out_sym);
    gemm3_kernel<<<dim3(16, 2), 256, 0, stream>>>(zact, W3, b3, v, out_v);
}